// TextLoss_12747462934889
// MI455X (gfx1250) — compile-verified
//
#include <hip/hip_runtime.h>
#include <hip/hip_bf16.h>
#include <math.h>

// ---------------- problem constants ----------------
#define B_    8
#define C_    4
#define H_    512
#define W_    512
#define HW_   (H_ * W_)
#define NBIN  1024
#define NROW  (B_ * C_)   // 32 (b, channel) rows

// ---------------- workspace layout (floats) ----------------
enum : int {
  O_NEGCNT = 0,                       // [32][1024] counts (as float)
  O_NEGSUM = NROW * NBIN,             // [32][1024] sums
  O_POSCNT = 2 * NROW * NBIN,         // [32]
  O_POSSUM = O_POSCNT + NROW,         // [32]
  O_BCE1   = O_POSSUM + NROW,         // scalar
  O_INTER1 = O_BCE1 + 1,              // [8]
  O_XSUM1  = O_INTER1 + B_,           // [8]
  O_TSUM1  = O_XSUM1 + B_,            // [8]
  O_BCE2   = O_TSUM1 + B_,            // scalar
  O_INTER2 = O_BCE2 + 1,              // [8]
  O_XSUM2  = O_INTER2 + B_,           // [8]
  O_TSUM2  = O_XSUM2 + B_,            // [8]
  WS_FLOATS = O_TSUM2 + B_            // 65650 floats ~ 257 KB
};

// ---------------- WMMA wave32 reduction ----------------
// V_WMMA_F32_16X16X4_F32 with B = ones:  D[m][n] = sum_k A[m][k].
// A layout (16x4 f32): lanes 0-15 supply K=0,1 ; lanes 16-31 supply K=2,3.
// Put v in A-vgpr0, 0 in A-vgpr1  ->  rowsum_m = v_m + v_{m+16}.
// D layout: lane n<16 holds rows 0..7 in 8 VGPRs, lane n>=16 rows 8..15.
// Sum of the 8 D VGPRs gives a half-wave sum; shfl_xor(16) completes it.
typedef __attribute__((ext_vector_type(2))) float v2f;
typedef __attribute__((ext_vector_type(8))) float v8f;

__device__ __forceinline__ float wave_sum_wmma(float v) {
  v2f a; a[0] = v;    a[1] = 0.0f;
  v2f o; o[0] = 1.0f; o[1] = 1.0f;
  v8f c = {};
  v8f d = __builtin_amdgcn_wmma_f32_16x16x4_f32(false, a, false, o,
                                                (short)0, c, false, false);
  float s = d[0] + d[1] + d[2] + d[3] + d[4] + d[5] + d[6] + d[7];
  s += __shfl_xor(s, 16, 32);
  return s;  // full wave32 sum, uniform across lanes
}

__device__ __forceinline__ float sigmoidf(float x) {
  return 1.0f / (1.0f + expf(-x));
}
__device__ __forceinline__ float bce_term(float x, float t) {
  return fmaxf(x, 0.0f) - x * t + log1pf(expf(-fabsf(x)));
}

// ---------------- kernel 0: zero workspace ----------------
__global__ void zero_ws_kernel(float* __restrict__ ws, int n) {
  for (int i = blockIdx.x * blockDim.x + threadIdx.x; i < n;
       i += gridDim.x * blockDim.x)
    ws[i] = 0.0f;
}

// ---------------- kernel 1: fused regression pass (+ seg2 on channel 3) ----
// grid (Gx, 8), block 256. One batch image per blockIdx.y.
// Reads inputs/tr_mask/train_mask/label exactly once.
__global__ __launch_bounds__(256)
void reg_seg2_kernel(const float* __restrict__ inputs,
                     const float* __restrict__ tr_mask,
                     const float* __restrict__ train_mask,
                     const float* __restrict__ label,
                     float* __restrict__ ws) {
  __shared__ unsigned int hc[C_ * NBIN];  // 16 KB
  __shared__ float        hs[C_ * NBIN];  // 16 KB

  const int b   = blockIdx.y;
  const int tid = threadIdx.x;
  const int lane = tid & 31;

  const float*  inp_b = inputs + (size_t)b * C_ * HW_;
  const float4* tr4   = (const float4*)tr_mask + (size_t)b * HW_;  // (b,h,w,4)
  const float*  tm_b  = train_mask + (size_t)b * HW_;
  const float*  lab_b = label + (size_t)b * HW_;

  for (int j = tid; j < C_ * NBIN; j += blockDim.x) { hc[j] = 0u; hs[j] = 0.0f; }
  __syncthreads();

  float pos_cnt[C_] = {0.f, 0.f, 0.f, 0.f};
  float pos_sum[C_] = {0.f, 0.f, 0.f, 0.f};
  float bce2 = 0.f, inter2 = 0.f, xs2 = 0.f, ts2 = 0.f;

  const int stride = gridDim.x * blockDim.x;
  for (int p = blockIdx.x * blockDim.x + tid; p < HW_; p += stride) {
    float  tm  = tm_b[p];
    float4 t4  = tr4[p];
    float  lab = lab_b[p];
#pragma unroll
    for (int i = 0; i < C_; ++i) {
      __builtin_prefetch(&inp_b[i * HW_ + p + stride], 0, 1);  // global_prefetch_b8
      float x   = inp_b[i * HW_ + p];
      float sg  = sigmoidf(x);
      float t   = (i == 0) ? t4.x : (i == 1) ? t4.y : (i == 2) ? t4.z : t4.w;
      float dd  = sg - t;
      float reg = dd * dd * tm;
      if (t >= 0.001f) {               // EPS_POS
        pos_cnt[i] += 1.0f;
        pos_sum[i] += reg;
      } else {
        int bin = (int)(reg * (float)NBIN);
        bin = bin > (NBIN - 1) ? (NBIN - 1) : (bin < 0 ? 0 : bin);
        atomicAdd(&hc[i * NBIN + bin], 1u);    // ds_add_u32
        atomicAdd(&hs[i * NBIN + bin], reg);   // ds_add_f32
      }
      if (i == C_ - 1) {               // fused seg2 on last channel
        bce2   += bce_term(x, lab);
        inter2 += sg * lab;
        xs2    += sg;
        ts2    += lab;
      }
    }
  }
  __syncthreads();

  // merge LDS histograms -> global (skip empty bins)
  const int row0 = b * C_;
  for (int j = tid; j < C_ * NBIN; j += blockDim.x) {
    unsigned cv = hc[j];
    float    sv = hs[j];
    int i = j >> 10, bin = j & (NBIN - 1);
    if (cv)          atomicAdd(&ws[O_NEGCNT + (row0 + i) * NBIN + bin], (float)cv);
    if (sv != 0.0f)  atomicAdd(&ws[O_NEGSUM + (row0 + i) * NBIN + bin], sv);
  }

  // wave-level WMMA reductions, one atomic per wave (EXEC all-ones here)
#pragma unroll
  for (int i = 0; i < C_; ++i) {
    float wc = wave_sum_wmma(pos_cnt[i]);
    float wsm = wave_sum_wmma(pos_sum[i]);
    if (lane == 0) {
      atomicAdd(&ws[O_POSCNT + row0 + i], wc);
      atomicAdd(&ws[O_POSSUM + row0 + i], wsm);
    }
  }
  float wb = wave_sum_wmma(bce2);
  float wi = wave_sum_wmma(inter2);
  float wx = wave_sum_wmma(xs2);
  float wt = wave_sum_wmma(ts2);
  if (lane == 0) {
    atomicAdd(&ws[O_BCE2],       wb);
    atomicAdd(&ws[O_INTER2 + b], wi);
    atomicAdd(&ws[O_XSUM2 + b],  wx);
    atomicAdd(&ws[O_TSUM2 + b],  wt);
  }
}

// ---------------- kernel 2: seg1 pass (us_input vs one-hot label) ----------
__global__ __launch_bounds__(256)
void seg1_kernel(const float* __restrict__ us,
                 const float* __restrict__ label,
                 float* __restrict__ ws) {
  const int b   = blockIdx.y;
  const int tid = threadIdx.x;
  const int lane = tid & 31;

  const float* u0    = us + (size_t)b * 2 * HW_;
  const float* u1    = u0 + HW_;
  const float* lab_b = label + (size_t)b * HW_;

  float bce = 0.f, inter = 0.f, xs = 0.f, ts = 0.f;
  const int stride = gridDim.x * blockDim.x;
  for (int p = blockIdx.x * blockDim.x + tid; p < HW_; p += stride) {
    float lab = lab_b[p];
    float t0 = 1.0f - lab, t1 = lab;    // (label==0), (label==1)
    float x0 = u0[p], x1 = u1[p];
    float s0 = sigmoidf(x0), s1 = sigmoidf(x1);
    bce   += bce_term(x0, t0) + bce_term(x1, t1);
    inter += s0 * t0 + s1 * t1;
    xs    += s0 + s1;
    ts    += t0 + t1;
  }
  float wb = wave_sum_wmma(bce);
  float wi = wave_sum_wmma(inter);
  float wx = wave_sum_wmma(xs);
  float wt = wave_sum_wmma(ts);
  if (lane == 0) {
    atomicAdd(&ws[O_BCE1],       wb);
    atomicAdd(&ws[O_INTER1 + b], wi);
    atomicAdd(&ws[O_XSUM1 + b],  wx);
    atomicAdd(&ws[O_TSUM1 + b],  wt);
  }
}

// ---------------- kernel 3: finalize (histogram top-k walk + scalar) -------
__device__ float hist_topk_sum(const float* __restrict__ ws, int r, float kf) {
  const float* hc = ws + O_NEGCNT + r * NBIN;
  const float* hsum = ws + O_NEGSUM + r * NBIN;
  float cnt = 0.f, sum = 0.f;
  for (int bin = NBIN - 1; bin >= 0; --bin) {
    float bc = hc[bin];
    if (cnt + bc >= kf) {
      float need = kf - cnt;
      if (bc > 0.f) sum += hsum[bin] * (need / bc);  // in-bin mean interpolation
      return sum;
    }
    cnt += bc;
    sum += hsum[bin];
  }
  return sum;
}

__global__ void finalize_kernel(const float* __restrict__ ws,
                                const float* __restrict__ sigw,
                                float* __restrict__ out) {
  __shared__ float items[NROW];
  const int r = threadIdx.x;
  if (r < NROW) {
    const float Nf = (float)HW_;
    float pc   = ws[O_POSCNT + r];
    float negc = Nf - pc;
    float kf   = fminf(negc, 3.0f * pc);
    float item;
    if (pc > 0.5f) {
      float posi = ws[O_POSSUM + r] / pc;
      float nega = 0.f;
      if (kf > 0.5f) nega = hist_topk_sum(ws, r, kf) / kf;
      item = posi + nega;
    } else {
      // pos_count==0 -> full-row hist == neg hist; top-100 mean
      item = hist_topk_sum(ws, r, 100.0f) * (1.0f / 100.0f);
    }
    items[r] = item;
  }
  __syncthreads();
  if (threadIdx.x == 0) {
    float rows = 0.f;
    for (int i = 0; i < NROW; ++i) rows += items[i];
    float loss = rows / (float)B_;
    float sw = sigw[0];
    loss = loss / (2.0f * sw * sw + 1e-6f) - logf(sw);

    const float S = 1e-5f;
    float bce1 = ws[O_BCE1] / (float)((size_t)B_ * 2 * HW_);
    float ds1 = 0.f;
    for (int b = 0; b < B_; ++b)
      ds1 += (2.0f * ws[O_INTER1 + b] + S) /
             (ws[O_XSUM1 + b] + ws[O_TSUM1 + b] + S);
    float seg1 = 0.5f * bce1 + (1.0f - ds1 / (float)B_);

    float bce2 = ws[O_BCE2] / (float)((size_t)B_ * HW_);
    float ds2 = 0.f;
    for (int b = 0; b < B_; ++b)
      ds2 += (2.0f * ws[O_INTER2 + b] + S) /
             (ws[O_XSUM2 + b] + ws[O_TSUM2 + b] + S);
    float seg2 = 0.5f * bce2 + (1.0f - ds2 / (float)B_);

    out[0] = seg1 + seg2 + loss;
  }
}

// ---------------- launcher ----------------
extern "C" void kernel_launch(void* const* d_in, const int* in_sizes, int n_in,
                              void* d_out, int out_size, void* d_ws, size_t ws_size,
                              hipStream_t stream) {
  const float* us_input   = (const float*)d_in[0];  // (8,2,512,512)
  const float* inputs     = (const float*)d_in[1];  // (8,4,512,512)
  const float* train_mask = (const float*)d_in[2];  // (8,1,512,512)
  const float* tr_mask    = (const float*)d_in[3];  // (8,512,512,4)
  const float* label      = (const float*)d_in[4];  // (8,1,512,512)
  const float* sig_weight = (const float*)d_in[5];  // (1,)
  float* out = (float*)d_out;
  float* ws  = (float*)d_ws;

  zero_ws_kernel<<<(WS_FLOATS + 255) / 256, 256, 0, stream>>>(ws, WS_FLOATS);

  dim3 g2(128, B_);
  reg_seg2_kernel<<<g2, 256, 0, stream>>>(inputs, tr_mask, train_mask, label, ws);

  dim3 g1(64, B_);
  seg1_kernel<<<g1, 256, 0, stream>>>(us_input, label, ws);

  finalize_kernel<<<1, 32, 0, stream>>>(ws, sig_weight, out);
}